// TensorDLT_31181462569251
// MI455X (gfx1250) — compile-verified
//
#include <hip/hip_runtime.h>

#define THREADS 256
#define IN_STRIDE 20   // floats per thread in sIn (80 B, 16B-aligned, cuts DS bank conflicts 8-way -> 2-way)

// Batched 4-point homography (TensorDLT).
// SVD null-vector == solution of A[:, :8] h = -A[:, 8] with h8 = 1 (since the
// reference normalizes by H[2,2]).  One thread per batch element, 8x8 Gaussian
// elimination fully unrolled in VGPRs.  Data movement uses the CDNA5 async
// global<->LDS path (ASYNCcnt) for both input staging and coalesced output.
__global__ __launch_bounds__(THREADS)
void tensor_dlt_kernel(const float* __restrict__ src,
                       const float* __restrict__ dst,
                       float* __restrict__ out,
                       int B)
{
    __shared__ float sIn[THREADS * IN_STRIDE];  // per thread: 8 src + 8 dst floats + pad (20 KB)
    __shared__ float sOut[THREADS * 9];         // per thread: 9 output floats (9 KB)

    const int t = threadIdx.x;
    int gid = blockIdx.x * THREADS + t;
    if (gid > B - 1) gid = B - 1;         // clamp (no divergence; B % 256 == 0 anyway)

    // ---- Async DMA inputs into LDS (global_load_async_to_lds_b128) ----
    // LDS layout: sIn[t*IN_STRIDE + 0..7] = src point data, + 8..15 = dst.
    const unsigned ldsSrc = (unsigned)(uintptr_t)&sIn[t * IN_STRIDE];  // low 32 bits of a
    const unsigned ldsDst = ldsSrc + 32u;                              // generic LDS ptr = LDS offset
    const unsigned goff   = (unsigned)gid * 32u;                       // byte offset (8 floats)

    asm volatile(
        "global_load_async_to_lds_b128 %0, %2, %3\n\t"
        "global_load_async_to_lds_b128 %0, %2, %3 offset:16\n\t"
        "global_load_async_to_lds_b128 %1, %2, %4\n\t"
        "global_load_async_to_lds_b128 %1, %2, %4 offset:16\n\t"
        "s_wait_asynccnt 0x0"
        :
        : "v"(ldsSrc), "v"(ldsDst), "v"(goff), "s"(src), "s"(dst)
        : "memory");

    // Each thread consumes only the LDS bytes its own lane requested, so the
    // per-wave s_wait_asynccnt above is sufficient (no barrier needed here).
    const float4 s0 = *(const float4*)&sIn[t * IN_STRIDE + 0];   // x0 y0 x1 y1
    const float4 s1 = *(const float4*)&sIn[t * IN_STRIDE + 4];   // x2 y2 x3 y3
    const float4 d0 = *(const float4*)&sIn[t * IN_STRIDE + 8];   // u0 v0 u1 v1
    const float4 d1 = *(const float4*)&sIn[t * IN_STRIDE + 12];  // u2 v2 u3 v3

    // ---- Build augmented 8x9 system:  A[:, :8] h = -A[:, 8]  (col 8 = u / v) ----
    float M[8][9];
    {
        const float xs[4] = {s0.x, s0.z, s1.x, s1.z};
        const float ys[4] = {s0.y, s0.w, s1.y, s1.w};
        const float us[4] = {d0.x, d0.z, d1.x, d1.z};
        const float vs[4] = {d0.y, d0.w, d1.y, d1.w};
#pragma unroll
        for (int i = 0; i < 4; ++i) {
            const float x = xs[i], y = ys[i], u = us[i], v = vs[i];
            float* r0 = M[2 * i + 0];
            float* r1 = M[2 * i + 1];
            r0[0] = x;    r0[1] = y;    r0[2] = 1.0f;
            r0[3] = 0.0f; r0[4] = 0.0f; r0[5] = 0.0f;
            r0[6] = -u * x; r0[7] = -u * y; r0[8] = u;
            r1[0] = 0.0f; r1[1] = 0.0f; r1[2] = 0.0f;
            r1[3] = x;    r1[4] = y;    r1[5] = 1.0f;
            r1[6] = -v * x; r1[7] = -v * y; r1[8] = v;
        }
    }

    // ---- Gaussian elimination, branch-free partial pivoting, fully unrolled ----
#pragma unroll
    for (int k = 0; k < 8; ++k) {
#pragma unroll
        for (int r = k + 1; r < 8; ++r) {
            const bool sw = fabsf(M[r][k]) > fabsf(M[k][k]);
#pragma unroll
            for (int c = k; c < 9; ++c) {
                const float a = M[k][c], b = M[r][c];
                M[k][c] = sw ? b : a;
                M[r][c] = sw ? a : b;
            }
        }
        const float inv = 1.0f / M[k][k];
#pragma unroll
        for (int r = k + 1; r < 8; ++r) {
            const float f = M[r][k] * inv;
#pragma unroll
            for (int c = k + 1; c < 9; ++c)
                M[r][c] = fmaf(-f, M[k][c], M[r][c]);
        }
    }

    // ---- Back substitution ----
    float h[8];
#pragma unroll
    for (int k = 7; k >= 0; --k) {
        float sacc = M[k][8];
#pragma unroll
        for (int c = k + 1; c < 8; ++c)
            sacc = fmaf(-M[k][c], h[c], sacc);
        h[k] = sacc / M[k][k];
    }

    // ---- Stage results in LDS, then async-store coalesced (4B per lane) ----
    // Stride 9 floats is coprime with the 64 banks -> conflict-free stores; the
    // transposed readback (idx = i*256 + t) is lane-consecutive -> conflict-free.
#pragma unroll
    for (int i = 0; i < 8; ++i) sOut[t * 9 + i] = h[i];
    sOut[t * 9 + 8] = 1.0f;   // H[2,2] == 1 after normalization

    __syncthreads();          // other lanes' LDS data feeds our async stores

    const unsigned sOutBase     = (unsigned)(uintptr_t)&sOut[0];
    const unsigned blockOutElem = (unsigned)blockIdx.x * (THREADS * 9u);
    const unsigned totalElem    = (unsigned)B * 9u;
#pragma unroll
    for (int i = 0; i < 9; ++i) {
        const unsigned idx = (unsigned)(i * THREADS) + (unsigned)t;
        const unsigned g   = blockOutElem + idx;      // flat output element
        if (g < totalElem) {
            const unsigned memOff = g * 4u;           // byte offset into out
            const unsigned ldsA   = sOutBase + idx * 4u;
            asm volatile("global_store_async_from_lds_b32 %0, %1, %2"
                         :
                         : "v"(memOff), "v"(ldsA), "s"(out)
                         : "memory");
        }
    }
    asm volatile("s_wait_asynccnt 0x0" ::: "memory");
}

extern "C" void kernel_launch(void* const* d_in, const int* in_sizes, int n_in,
                              void* d_out, int out_size, void* d_ws, size_t ws_size,
                              hipStream_t stream) {
    const float* src = (const float*)d_in[0];   // (B, 8) float32
    const float* dst = (const float*)d_in[1];   // (B, 8) float32
    float* out = (float*)d_out;                 // (B, 3, 3) float32
    const int B = in_sizes[0] / 8;
    const int grid = (B + THREADS - 1) / THREADS;
    tensor_dlt_kernel<<<grid, THREADS, 0, stream>>>(src, dst, out, B);
}